// WCT_29489245454770
// MI455X (gfx1250) — compile-verified
//
#include <hip/hip_runtime.h>
#include <math.h>

#define ALPHA 0.4f
#define EPS   1e-5f
#define SLOPE 0.01f
#define HW    16384
#define NTOTF 65536.0f

typedef __attribute__((ext_vector_type(2))) float v2f;
typedef __attribute__((ext_vector_type(8))) float v8f;

// D(16x16,f32) = A(16x4,f32) * B(4x16,f32) + C
__device__ __forceinline__ v8f wmma4(v2f a, v2f b, v8f c) {
  return __builtin_amdgcn_wmma_f32_16x16x4_f32(false, a, false, b, (short)0, c,
                                               false, false);
}

// ---- CDNA5 async global->LDS path (guarded; falls back to sync staging) ----
#if __has_builtin(__builtin_amdgcn_global_load_async_to_lds_b128) && \
    __has_builtin(__builtin_amdgcn_s_wait_asynccnt)
#define ASYNC_LDS 1
#else
#define ASYNC_LDS 0
#endif

#if ASYNC_LDS
typedef int v4i __attribute__((vector_size(4 * sizeof(int))));
typedef __attribute__((address_space(1))) v4i as1_v4i;
typedef __attribute__((address_space(3))) v4i as3_v4i;
#endif

__device__ __forceinline__ void async_copy_b128(const float* gsrc, float* ldst) {
#if ASYNC_LDS
  // generic shared pointer low 32 bits == wave-relative LDS byte offset
  __builtin_amdgcn_global_load_async_to_lds_b128(
      (as1_v4i*)(uintptr_t)gsrc, (as3_v4i*)(uintptr_t)ldst, 0, 0);
#else
  (void)gsrc; (void)ldst;
#endif
}

__device__ __forceinline__ void wait_async_lds() {
#if ASYNC_LDS
  __builtin_amdgcn_s_wait_asynccnt(0);
#endif
}

// ---------------------------------------------------------------- zero scratch
__global__ void wct_zero(float* __restrict__ ws, int n) {
  int i = blockIdx.x * 256 + threadIdx.x;
  if (i < n) ws[i] = 0.f;
}

// ---------------------------------------------------------------- MLP helpers
__device__ void layerf(const float* __restrict__ W, const float* __restrict__ bias,
                       const float* __restrict__ xin, float* __restrict__ xout,
                       int in_dim, int out_dim, bool relu) {
  for (int o = threadIdx.x; o < out_dim; o += blockDim.x) {
    float acc = bias[o];
    const float* wr = W + o * in_dim;
    for (int i = 0; i < in_dim; ++i) acc = fmaf(xin[i], wr[i], acc);
    xout[o] = relu ? ((acc >= 0.f) ? acc : SLOPE * acc) : acc;
  }
}

// blocks 0..31: ct-MLP row r (b=r/8, g=r%8) -> U, eigen_s, M' = a*UDU+(1-a)I
// blocks 32..35: mu-MLP row b -> s_mu
__global__ void wct_mlp(const float* __restrict__ sB,
                        const float* w0, const float* b0, const float* w1, const float* b1,
                        const float* w2, const float* b2, const float* w3, const float* b3,
                        const float* w4, const float* b4,
                        const float* mw0, const float* mb0, const float* mw1, const float* mb1,
                        const float* mw2, const float* mb2,
                        float* __restrict__ outEig, float* __restrict__ wsM,
                        float* __restrict__ wsSmu) {
  __shared__ float bufA[1040];
  __shared__ float bufB[1040];
  __shared__ float Ubuf[1024];
  __shared__ float Dbuf[32];
  const int t = threadIdx.x;
  const int r = blockIdx.x;
  if (r < 32) {
    if (t < 32) bufA[t] = sB[r * 32 + t];
    __syncthreads();
    layerf(w0, b0, bufA, bufB, 32, 256, true);   __syncthreads();
    layerf(w1, b1, bufB, bufA, 256, 256, true);  __syncthreads();
    layerf(w2, b2, bufA, bufB, 256, 256, true);  __syncthreads();
    layerf(w3, b3, bufB, bufA, 256, 256, true);  __syncthreads();
    layerf(w4, b4, bufA, bufB, 256, 1024, false); __syncthreads();
    // s_CT (32x32) now in bufB, row-major [i*32+j]
    if (t < 32) {                      // column norms D[j]
      float s = 0.f;
      for (int i = 0; i < 32; ++i) { float v = bufB[i * 32 + t]; s += v * v; }
      Dbuf[t] = sqrtf(s);
    }
    __syncthreads();
    const int b = r >> 3, g = r & 7;
    float* eig = outEig + (g * 4 + b) * 1024;   // eigen_s index g*B+b
    for (int e = t; e < 1024; e += 256) {
      float u = bufB[e] / Dbuf[e & 31];
      Ubuf[e] = u;
      eig[e] = u;
    }
    __syncthreads();
    for (int e = t; e < 1024; e += 256) {       // UDU[i][k] = sum_j sCT[i][j]*U[k][j]
      const int i = e >> 5, k = e & 31;
      float s = 0.f;
      for (int j = 0; j < 32; ++j) s = fmaf(bufB[i * 32 + j], Ubuf[k * 32 + j], s);
      wsM[r * 1024 + e] = ALPHA * s + ((i == k) ? (1.f - ALPHA) : 0.f);
    }
  } else {
    const int bb = r - 32;
    bufA[t] = sB[bb * 256 + t];
    __syncthreads();
    layerf(mw0, mb0, bufA, bufB, 256, 256, true);  __syncthreads();
    layerf(mw1, mb1, bufB, bufA, 256, 256, true);  __syncthreads();
    layerf(mw2, mb2, bufA, bufB, 256, 256, false); __syncthreads();
    wsSmu[bb * 256 + t] = bufB[t];
  }
}

// ---------------------------------------------------------------- stats pass
// 256 blocks x 128 thr: block = ((b*8+g)*8 + chunk). Accumulates per-channel
// sums S and per-group P = X*X^T via f32 WMMA. Sub-tiles are double-buffered
// through LDS with GLOBAL_LOAD_ASYNC_TO_LDS_B128 (ASYNCcnt) when available.
__global__ void wct_stats(const float* __restrict__ cA,
                          float* __restrict__ wsS, float* __restrict__ wsP) {
  __shared__ float tile[2][32 * 257];
  __shared__ float ssum[32][4];
  const int tid = threadIdx.x;
  const int bg = blockIdx.x >> 3;
  const int chunk = blockIdx.x & 7;
  const int b = bg >> 3, g = bg & 7;
  const float* X = cA + (size_t)(b * 256 + g * 32) * HW + chunk * 2048;
  const int w = tid >> 5, lane = tid & 31;
  const bool lh = lane < 16;
  const int rowq = lane & 15;
  const int koff = lh ? 0 : 2;
  const int mi = w >> 1, mj = w & 1;       // 4 waves -> 2x2 16x16 tiles of P
  v8f acc = {0.f, 0.f, 0.f, 0.f, 0.f, 0.f, 0.f, 0.f};
  float Sacc = 0.f;

  auto stage = [&](int sub, int buf) {
    const int base = sub * 256;
#if ASYNC_LDS
    for (int idx = tid; idx < 2048; idx += 128) {      // 2048 x 16B transfers
      const int ch = idx >> 6, s4 = (idx & 63) << 2;
      async_copy_b128(&X[(size_t)ch * HW + base + s4], &tile[buf][ch * 257 + s4]);
    }
#else
    for (int idx = tid; idx < 32 * 256; idx += 128) {
      const int ch = idx >> 8, s = idx & 255;
      tile[buf][ch * 257 + s] = X[(size_t)ch * HW + base + s];
    }
#endif
  };

  stage(0, 0);
  wait_async_lds();
  __syncthreads();

  for (int sub = 0; sub < 8; ++sub) {
    const int cur = sub & 1;
    if (sub < 7) stage(sub + 1, cur ^ 1);   // prefetch next sub-tile
    const float* tl = tile[cur];
    { // per-channel partial sums
      const int ch = tid >> 2, q = tid & 3;
      float p = 0.f;
      const float* row = &tl[ch * 257 + q * 64];
      for (int s = 0; s < 64; ++s) p += row[s];
      ssum[ch][q] = p;
    }
    // P-tile accumulation: K = 256 samples in steps of 4
    const int ca = mi * 16 + rowq;           // A-fragment channel
    const int cb = mj * 16 + rowq;           // B-fragment channel
    #pragma unroll 4
    for (int k0 = 0; k0 < 256; k0 += 4) {
      v2f a, bf;
      a.x  = tl[ca * 257 + k0 + koff];
      a.y  = tl[ca * 257 + k0 + koff + 1];
      bf.x = tl[cb * 257 + k0 + koff];
      bf.y = tl[cb * 257 + k0 + koff + 1];
      acc = wmma4(a, bf, acc);
    }
    __syncthreads();
    if (tid < 32)
      Sacc += ssum[tid][0] + ssum[tid][1] + ssum[tid][2] + ssum[tid][3];
    if (sub < 7) wait_async_lds();          // next buffer complete (this wave)
    __syncthreads();                         // ... and for all waves
  }
  #pragma unroll
  for (int vv = 0; vv < 8; ++vv) {
    const int i = mi * 16 + vv + (lh ? 0 : 8);
    const int j = mj * 16 + rowq;
    atomicAdd(&wsP[g * 1024 + i * 32 + j], acc[vv]);
  }
  if (tid < 32) atomicAdd(&wsS[g * 32 + tid], Sacc);
}

// ---------------------------------------------------------------- finalize
// 8 blocks x 256 thr: cov_c out, mean, fused bias = (1-a)*mean + a*s_mu
__global__ void wct_finalize(const float* __restrict__ wsS, const float* __restrict__ wsP,
                             const float* __restrict__ wsSmu, float* __restrict__ wsMean,
                             float* __restrict__ wsBias, float* __restrict__ outCov) {
  const int g = blockIdx.x, t = threadIdx.x;
  const float invN = 1.f / NTOTF;
  const float invN1 = 1.f / (NTOTF - 1.f);
  for (int e = t; e < 1024; e += 256) {
    const int i = e >> 5, j = e & 31;
    const float Si = wsS[g * 32 + i], Sj = wsS[g * 32 + j];
    outCov[g * 1024 + e] =
        (wsP[g * 1024 + e] - Si * Sj * invN) * invN1 + ((i == j) ? EPS : 0.f);
  }
  if (t < 32) wsMean[g * 32 + t] = wsS[g * 32 + t] * invN;
  if (t < 128) {
    const int b = t >> 5, i = t & 31, c = g * 32 + i;
    wsBias[b * 256 + c] = (1.f - ALPHA) * (wsS[c] * invN) + ALPHA * wsSmu[b * 256 + c];
  }
}

// ---------------------------------------------------------------- coloring GEMM
// 256 blocks x 256 thr: out[b, g*32+c, n] = sum_d M'[c,d]*(cA[b,g*32+d,n]-mean) + bias
// A-fragments (M', 32x32) and bias live in registers; accumulators start at bias.
__global__ void wct_color(const float* __restrict__ cA, const float* __restrict__ wsM,
                          const float* __restrict__ wsMean, const float* __restrict__ wsBias,
                          float* __restrict__ out) {
  const int tid = threadIdx.x;
  const int bg = blockIdx.x >> 3;
  const int chunk = blockIdx.x & 7;
  const int b = bg >> 3, g = bg & 7;
  const int w = tid >> 5, lane = tid & 31;
  const bool lh = lane < 16;
  const int rowq = lane & 15;
  const int koff = lh ? 0 : 2;
  const int rowoff = lh ? 0 : 8;
  const float* Mp = wsM + bg * 1024;
  v2f Af[2][8];
  #pragma unroll
  for (int mt = 0; mt < 2; ++mt) {
    const int rr = mt * 16 + rowq;
    #pragma unroll
    for (int ks = 0; ks < 8; ++ks) {
      Af[mt][ks].x = Mp[rr * 32 + ks * 4 + koff];
      Af[mt][ks].y = Mp[rr * 32 + ks * 4 + koff + 1];
    }
  }
  float bias0[8], bias1[8];
  #pragma unroll
  for (int vv = 0; vv < 8; ++vv) {
    bias0[vv] = wsBias[b * 256 + g * 32 + vv + rowoff];
    bias1[vv] = wsBias[b * 256 + g * 32 + 16 + vv + rowoff];
  }
  float m0[8], m1[8];
  #pragma unroll
  for (int ks = 0; ks < 8; ++ks) {
    m0[ks] = wsMean[g * 32 + ks * 4 + koff];
    m1[ks] = wsMean[g * 32 + ks * 4 + koff + 1];
  }
  const float* Xb = cA + (size_t)(b * 256 + g * 32) * HW;
  float* Ob = out + (size_t)(b * 256 + g * 32) * HW;
  const int nbase = chunk * 2048 + w * 256;
  for (int tt = 0; tt < 16; ++tt) {
    const int col = nbase + tt * 16 + rowq;
    __builtin_prefetch(Xb + col + 16, 0, 0);
    v8f acc0, acc1;
    #pragma unroll
    for (int vv = 0; vv < 8; ++vv) { acc0[vv] = bias0[vv]; acc1[vv] = bias1[vv]; }
    #pragma unroll
    for (int ks = 0; ks < 8; ++ks) {
      const int ch = ks * 4 + koff;
      v2f bf;
      bf.x = Xb[(size_t)ch * HW + col] - m0[ks];
      bf.y = Xb[(size_t)(ch + 1) * HW + col] - m1[ks];
      acc0 = wmma4(Af[0][ks], bf, acc0);
      acc1 = wmma4(Af[1][ks], bf, acc1);
    }
    #pragma unroll
    for (int vv = 0; vv < 8; ++vv) {
      __builtin_nontemporal_store(acc0[vv], &Ob[(size_t)(vv + rowoff) * HW + col]);
      __builtin_nontemporal_store(acc1[vv], &Ob[(size_t)(16 + vv + rowoff) * HW + col]);
    }
  }
}

// ---------------------------------------------------------------- launcher
extern "C" void kernel_launch(void* const* d_in, const int* in_sizes, int n_in,
                              void* d_out, int out_size, void* d_ws, size_t ws_size,
                              hipStream_t stream) {
  const float* cA = (const float*)d_in[0];
  const float* sB = (const float*)d_in[1];
  // d_in[2] (mask) unused: block-diagonal structure exploited analytically
  const float* ctw0 = (const float*)d_in[3];  const float* ctb0 = (const float*)d_in[4];
  const float* ctw1 = (const float*)d_in[5];  const float* ctb1 = (const float*)d_in[6];
  const float* ctw2 = (const float*)d_in[7];  const float* ctb2 = (const float*)d_in[8];
  const float* ctw3 = (const float*)d_in[9];  const float* ctb3 = (const float*)d_in[10];
  const float* ctw4 = (const float*)d_in[11]; const float* ctb4 = (const float*)d_in[12];
  const float* muw0 = (const float*)d_in[13]; const float* mub0 = (const float*)d_in[14];
  const float* muw1 = (const float*)d_in[15]; const float* mub1 = (const float*)d_in[16];
  const float* muw2 = (const float*)d_in[17]; const float* mub2 = (const float*)d_in[18];

  float* out = (float*)d_out;
  float* ws = (float*)d_ws;
  // workspace layout (floats)
  float* wsS    = ws;            // 256   per-channel sums (zeroed)
  float* wsP    = ws + 256;      // 8192  per-group X*X^T (zeroed)
  float* wsM    = ws + 8448;     // 32768 M' = a*UDU + (1-a)I per (b,g)
  float* wsSmu  = ws + 41216;    // 1024  s_mu
  float* wsBias = ws + 42240;    // 1024  fused bias
  float* wsMean = ws + 43264;    // 256   per-channel mean
  // output layout: out[16777216] | cov_c[8192] | eigen_s[32768]
  float* outCov = out + 16777216;
  float* outEig = out + 16785408;

  hipLaunchKernelGGL(wct_zero, dim3(33), dim3(256), 0, stream, ws, 8448);
  hipLaunchKernelGGL(wct_mlp, dim3(36), dim3(256), 0, stream, sB,
                     ctw0, ctb0, ctw1, ctb1, ctw2, ctb2, ctw3, ctb3, ctw4, ctb4,
                     muw0, mub0, muw1, mub1, muw2, mub2, outEig, wsM, wsSmu);
  hipLaunchKernelGGL(wct_stats, dim3(256), dim3(128), 0, stream, cA, wsS, wsP);
  hipLaunchKernelGGL(wct_finalize, dim3(8), dim3(256), 0, stream, wsS, wsP, wsSmu,
                     wsMean, wsBias, outCov);
  hipLaunchKernelGGL(wct_color, dim3(256), dim3(256), 0, stream, cA, wsM, wsMean,
                     wsBias, out);
}